// TopologyLoss_29867202576958
// MI455X (gfx1250) — compile-verified
//
#include <hip/hip_runtime.h>
#include <math.h>

// ---------------------------------------------------------------------------
// clDice topology loss, MI455X (gfx1250).
// Bandwidth-bound morphological stencil: no MAC structure -> WMMA does not
// apply. The CDNA5 win is the async global->LDS copy path + fused
// erode/dilate/skel-update (one pass per skeletonization iteration).
// ---------------------------------------------------------------------------

#define IMG   1024
#define TILE  32
#define SW    40            // s_img row stride (36 cols, padded)
#define EW    36            // s_ero row stride (34 cols)
#define NPIX  (IMG * IMG)   // per image
#define NIMG  32
#define EPSL  1e-6f

typedef long long i64;

#if defined(__gfx1250__) && __has_builtin(__builtin_amdgcn_global_load_async_to_lds_b32)
#define HAVE_ASYNC_LDS 1
typedef __attribute__((address_space(1))) int gas_int;
typedef __attribute__((address_space(3))) int las_int;
#define ASYNC_B32(gp, lp)                                                     \
  __builtin_amdgcn_global_load_async_to_lds_b32(                              \
      (gas_int*)(void*)(gp), (las_int*)(void*)(lp), 0, 0)
#if __has_builtin(__builtin_amdgcn_s_wait_asynccnt)
#define WAIT_ASYNC() __builtin_amdgcn_s_wait_asynccnt(0)
#else
#define WAIT_ASYNC() asm volatile("s_wait_asynccnt 0" ::: "memory")
#endif
#endif

__device__ __forceinline__ float reluf(float x) { return fmaxf(x, 0.0f); }
__device__ __forceinline__ float sigf(float x) { return 1.0f / (1.0f + expf(-x)); }

// ---------------------------------------------------------------------------
// init: p = sigmoid(pred); g = gt; skel_p = skel_g = 0.   (float4 vectorized)
// ---------------------------------------------------------------------------
__global__ __launch_bounds__(256) void cl_init(const float4* __restrict__ pred,
                                               const float4* __restrict__ gt,
                                               float4* __restrict__ p0,
                                               float4* __restrict__ g0,
                                               float4* __restrict__ sp,
                                               float4* __restrict__ sg, i64 n4) {
  i64 i = (i64)blockIdx.x * blockDim.x + threadIdx.x;
  i64 stride = (i64)gridDim.x * blockDim.x;
  const float4 z = make_float4(0.f, 0.f, 0.f, 0.f);
  for (; i < n4; i += stride) {
    float4 x = pred[i];
    float4 r;
    r.x = sigf(x.x); r.y = sigf(x.y); r.z = sigf(x.z); r.w = sigf(x.w);
    p0[i] = r;
    g0[i] = gt[i];
    sp[i] = z;
    sg[i] = z;
  }
}

// ---------------------------------------------------------------------------
// One soft-skeletonization iteration, fused:
//   eroded = erode3x3(img); opened = dilate3x3(eroded);
//   skel += relu(relu(img - opened) - skel); img_next = eroded.
// 32x32 output tile, 36x36 staged halo (via async global->LDS on gfx1250).
// blockIdx.z in [0,64): z<32 -> pred tensor, else gt tensor.
// ---------------------------------------------------------------------------
__global__ __launch_bounds__(256) void cl_skel_iter(
    const float* __restrict__ srcP, float* __restrict__ dstP, float* __restrict__ skelP,
    const float* __restrict__ srcG, float* __restrict__ dstG, float* __restrict__ skelG) {
  __shared__ float s_img[36 * SW];
  __shared__ float s_ero[34 * EW];

  const int z = blockIdx.z;
  const float* src; float* dst; float* skel; int img;
  if (z < NIMG) { src = srcP; dst = dstP; skel = skelP; img = z; }
  else          { src = srcG; dst = dstG; skel = skelG; img = z - NIMG; }
  const i64 base = (i64)img * NPIX;
  const int bx = blockIdx.x * TILE;
  const int by = blockIdx.y * TILE;
  const int t  = threadIdx.x;

  // ---- stage img tile + 2-halo; out-of-image = +INF (erode neutral) ----
  for (int i = t; i < 36 * 36; i += 256) {
    const int r = i / 36, c = i % 36;
    const int gy = by + r - 2, gx = bx + c - 2;
    const bool inb = (gy >= 0) & (gy < IMG) & (gx >= 0) & (gx < IMG);
#ifdef HAVE_ASYNC_LDS
    if (inb) {
      ASYNC_B32(src + base + (i64)gy * IMG + gx, &s_img[r * SW + c]);
    } else {
      s_img[r * SW + c] = __builtin_inff();
    }
#else
    s_img[r * SW + c] = inb ? src[base + (i64)gy * IMG + gx] : __builtin_inff();
#endif
  }
#ifdef HAVE_ASYNC_LDS
  WAIT_ASYNC();
#endif
  // hint L2 for the skel lines this block will read+write
  {
    const int tx = t & 31, ty = t >> 5;
    __builtin_prefetch(&skel[base + (i64)(by + ty) * IMG + (bx + tx)], 1, 0);
  }
  __syncthreads();

  // ---- eroded on tile + 1-halo; positions outside image = -INF (dilate pad)
  for (int i = t; i < 34 * 34; i += 256) {
    const int r = i / 34, c = i % 34;
    const int gy = by + r - 1, gx = bx + c - 1;
    float m = __builtin_inff();
#pragma unroll
    for (int dy = 0; dy < 3; ++dy)
#pragma unroll
      for (int dx = 0; dx < 3; ++dx)
        m = fminf(m, s_img[(r + dy) * SW + (c + dx)]);
    const bool inb = (gy >= 0) & (gy < IMG) & (gx >= 0) & (gx < IMG);
    s_ero[r * EW + c] = inb ? m : -__builtin_inff();
  }
  __syncthreads();

  // ---- per-pixel: opened, delta, skel update, write eroded as next img ----
  const int tx = t & 31, ty = t >> 5;
#pragma unroll
  for (int k = 0; k < 4; ++k) {
    const int r = ty + k * 8, c = tx;
    float opened = -__builtin_inff();
#pragma unroll
    for (int dy = 0; dy < 3; ++dy)
#pragma unroll
      for (int dx = 0; dx < 3; ++dx)
        opened = fmaxf(opened, s_ero[(r + dy) * EW + (c + dx)]);
    const float imgv = s_img[(r + 2) * SW + (c + 2)];
    const float ero  = s_ero[(r + 1) * EW + (c + 1)];
    const i64 g = base + (i64)(by + r) * IMG + (bx + c);
    const float delta = reluf(imgv - opened);
    const float s = skel[g];
    skel[g] = s + reluf(delta - s);
    dst[g] = ero;
  }
}

// ---------------------------------------------------------------------------
// Per-image reductions:
//   accum[img] = { sum(skel_p*gt), sum(skel_p), sum(skel_g*p), sum(skel_g) }
// grid: (1024, 1, 32); each block handles 1024 consecutive pixels (float4).
// ---------------------------------------------------------------------------
__global__ __launch_bounds__(256) void cl_reduce(const float4* __restrict__ pred,
                                                 const float4* __restrict__ gt,
                                                 const float4* __restrict__ skelP,
                                                 const float4* __restrict__ skelG,
                                                 float* __restrict__ accum) {
  const int img = blockIdx.z;
  const int t = threadIdx.x;
  const i64 i = (i64)img * (NPIX / 4) + (i64)blockIdx.x * 256 + t;

  const float4 pv = pred[i];
  const float4 gv = gt[i];
  const float4 sp = skelP[i];
  const float4 sg = skelG[i];
  const float px = sigf(pv.x), py = sigf(pv.y), pz = sigf(pv.z), pw = sigf(pv.w);

  float s1 = sp.x * gv.x + sp.y * gv.y + sp.z * gv.z + sp.w * gv.w;
  float s2 = sp.x + sp.y + sp.z + sp.w;
  float s3 = sg.x * px + sg.y * py + sg.z * pz + sg.w * pw;
  float s4 = sg.x + sg.y + sg.z + sg.w;

  // wave32 butterfly reduce
  for (int off = 16; off > 0; off >>= 1) {
    s1 += __shfl_down(s1, off);
    s2 += __shfl_down(s2, off);
    s3 += __shfl_down(s3, off);
    s4 += __shfl_down(s4, off);
  }
  __shared__ float red[8][4];
  const int lane = t & 31, wave = t >> 5;
  if (lane == 0) { red[wave][0] = s1; red[wave][1] = s2; red[wave][2] = s3; red[wave][3] = s4; }
  __syncthreads();
  if (t == 0) {
    float a0 = 0.f, a1 = 0.f, a2 = 0.f, a3 = 0.f;
#pragma unroll
    for (int w = 0; w < 8; ++w) { a0 += red[w][0]; a1 += red[w][1]; a2 += red[w][2]; a3 += red[w][3]; }
    atomicAdd(&accum[img * 4 + 0], a0);
    atomicAdd(&accum[img * 4 + 1], a1);
    atomicAdd(&accum[img * 4 + 2], a2);
    atomicAdd(&accum[img * 4 + 3], a3);
  }
}

__global__ void cl_finalize(const float* __restrict__ accum, float* __restrict__ out) {
  if (threadIdx.x == 0 && blockIdx.x == 0) {
    float acc = 0.f;
    for (int i = 0; i < NIMG; ++i) {
      const float s1 = accum[4 * i + 0], s2 = accum[4 * i + 1];
      const float s3 = accum[4 * i + 2], s4 = accum[4 * i + 3];
      const float tprec = s1 / (s2 + EPSL);
      const float tsens = s3 / (s4 + EPSL);
      acc += 2.f * tprec * tsens / (tprec + tsens + EPSL);
    }
    out[0] = 1.f - acc / (float)NIMG;
  }
}

// ---------------------------------------------------------------------------
extern "C" void kernel_launch(void* const* d_in, const int* in_sizes, int n_in,
                              void* d_out, int out_size, void* d_ws, size_t ws_size,
                              hipStream_t stream) {
  (void)in_sizes; (void)n_in; (void)out_size; (void)ws_size;
  const float* pred = (const float*)d_in[0];
  const float* gt   = (const float*)d_in[1];
  float* out = (float*)d_out;

  const i64 N = (i64)NIMG * NPIX;  // 33,554,432 elems per tensor
  float* p_a = (float*)d_ws;
  float* p_b = p_a + N;
  float* g_a = p_b + N;
  float* g_b = g_a + N;
  float* sp  = g_b + N;
  float* sg  = sp + N;
  float* accum = sg + N;  // 128 floats

  (void)hipMemsetAsync(accum, 0, NIMG * 4 * sizeof(float), stream);

  cl_init<<<2048, 256, 0, stream>>>((const float4*)pred, (const float4*)gt,
                                    (float4*)p_a, (float4*)g_a,
                                    (float4*)sp, (float4*)sg, N / 4);

  const float* srcP = p_a; float* dstP = p_b;
  const float* srcG = g_a; float* dstG = g_b;
  const dim3 grid(IMG / TILE, IMG / TILE, 2 * NIMG);
  for (int it = 0; it < 10; ++it) {
    cl_skel_iter<<<grid, 256, 0, stream>>>(srcP, dstP, sp, srcG, dstG, sg);
    float* tp = (float*)srcP; srcP = dstP; dstP = tp;
    float* tg = (float*)srcG; srcG = dstG; dstG = tg;
  }

  cl_reduce<<<dim3(NPIX / 1024, 1, NIMG), 256, 0, stream>>>(
      (const float4*)pred, (const float4*)gt,
      (const float4*)sp, (const float4*)sg, accum);

  cl_finalize<<<1, 32, 0, stream>>>(accum, out);
}